// YOLOv3Loss_44478681318144
// MI455X (gfx1250) — compile-verified
//
#include <hip/hip_runtime.h>

// ---------------------------------------------------------------------------
// YOLOv3 loss on gfx1250 (CDNA5). Sparse formulation: only target-marked cells
// contribute, so total work is ~2880 cells x 85 floats. Two tiny kernels,
// deterministic (no float atomics). Cross-lane sums use V_WMMA_F32_16X16X4_F32
// as a ones-matmul reducer (wave32).
// ---------------------------------------------------------------------------

typedef __attribute__((ext_vector_type(2))) float v2f;
typedef __attribute__((ext_vector_type(8))) float v8f;

// Sum one float per lane across the full wave32, result broadcast to all lanes.
// A[m][k]: lanes 0-15 vgpr0 -> A[m][0], lanes 16-31 vgpr0 -> A[m][2] (ISA 7.12.2).
// With B = ones(4x16): D[m][n] = x_m + x_{m+16}. d0..d7 hold rows {j, j+8}
// (lanes 0-15 / 16-31), so per-lane sum of d0..d7 gives the two half-sums;
// one xor-16 shuffle finishes.
__device__ __forceinline__ float wave_sum32(float x) {
  v2f a; a.x = x;    a.y = 0.0f;
  v2f b; b.x = 1.0f; b.y = 1.0f;
  v8f c = {};
  v8f d = __builtin_amdgcn_wmma_f32_16x16x4_f32(
      /*neg_a=*/false, a, /*neg_b=*/false, b,
      /*c_mod=*/(short)0, c, /*reuse_a=*/false, /*reuse_b=*/false);
  float t = ((d[0] + d[1]) + (d[2] + d[3])) + ((d[4] + d[5]) + (d[6] + d[7]));
  t += __shfl_xor(t, 16, 32);
  return t;
}

// One block per (batch, scale); 32 threads (one wave). Lane t handles target t.
__global__ __launch_bounds__(32) void yolo_scale_partial(
    const float* __restrict__ out0, const float* __restrict__ out1,
    const float* __restrict__ out2, const float* __restrict__ anc0,
    const float* __restrict__ anc1, const float* __restrict__ anc2,
    const float* __restrict__ targets, float* __restrict__ partial) {
  const int b = blockIdx.x;   // batch 0..31
  const int s = blockIdx.y;   // scale 0..2
  const int t = threadIdx.x;  // lane 0..31 (targets 0..29)

  const int g = (s == 0) ? 13 : (s == 1) ? 26 : 52;
  const float* out = (s == 0) ? out0 : (s == 1) ? out1 : out2;
  const float* anc = (s == 0) ? anc0 : (s == 1) ? anc1 : anc2;

  __shared__ int sCell[32];
  __shared__ int sCls[32];

  float tx = 0.f, ty = 0.f, tw = 0.f, th = 0.f;
  int cellIdx = -1;
  int cls = 0;

  if (t < 30) {
    const float* tg = targets + ((size_t)b * 30 + t) * 5;
    const float t0 = tg[0], t1 = tg[1], t2 = tg[2], t3 = tg[3];
    cls = (int)tg[4];

    const float gf = (float)g;
    const float gx = t0 * gf;
    const float gy = t1 * gf;
    const float sc = 416.0f * gf;              // IMG_DIM * g
    const float gw = fabsf(t2 - t0) * sc;
    const float gh = fabsf(t3 - t1) * sc;

    int best = 0;
    float bestIou = -1.0f;
#pragma unroll
    for (int a = 0; a < 3; ++a) {
      const float aw = anc[2 * a], ah = anc[2 * a + 1];
      const float inter = fminf(aw, gw) * fminf(ah, gh);
      const float uni = 1e-8f + aw * ah + gw * gh - inter;
      const float iou = inter / uni;
      if (iou > bestIou) { bestIou = iou; best = a; }  // first max wins (argmax)
    }

    const int gi = (int)gx;                    // trunc == floor (gx >= 0)
    const int gj = (int)gy;
    tx = gx - floorf(gx);
    ty = gy - floorf(gy);
    tw = logf(gw / anc[2 * best]);
    th = logf(gh / anc[2 * best + 1]);
    cellIdx = (best * g + gi) * g + gj;        // cell within this batch
  }

  sCell[t] = (t < 30) ? cellIdx : -1;
  sCls[t]  = cls;
  __syncthreads();

  // Last-write-wins dedup: lane t owns its cell iff no later target hits it.
  bool valid = (t < 30);
  if (valid) {
    for (int j = t + 1; j < 30; ++j)
      if (sCell[j] == cellIdx) { valid = false; break; }
  }

  float S = 0.0f, C = 0.0f, cnt = 0.0f;
  if (valid) {
    cnt = 1.0f;
    // tc is scattered per-class: the union of classes of ALL targets mapping
    // to this cell gets 1.0 (distinct 5-D indices -> no duplicate scatter).
    unsigned mw0 = 0, mw1 = 0, mw2 = 0;
    for (int j = 0; j < 30; ++j) {
      if (sCell[j] == cellIdx) {
        const int cj = sCls[j];
        if (cj < 32)      mw0 |= 1u << cj;
        else if (cj < 64) mw1 |= 1u << (cj - 32);
        else              mw2 |= 1u << (cj - 64);
      }
    }

    const float* o = out + ((size_t)b * 3 * g * g + cellIdx) * 85;
    const float d0 = o[0] - tx, d1 = o[1] - ty, d2 = o[2] - tw, d3 = o[3] - th;
    S = d0 * d0 + d1 * d1 + d2 * d2 + d3 * d3;

    const float LO = 1e-7f, HI = 1.0f - 1e-7f;
    for (int c = 0; c < 80; ++c) {
      float p = o[5 + c];
      p = fminf(fmaxf(p, LO), HI);
      const unsigned bit =
          (c < 32) ? (mw0 >> c) : (c < 64) ? (mw1 >> (c - 32)) : (mw2 >> (c - 64));
      C += (bit & 1u) ? -logf(p) : -logf(1.0f - p);
    }
  }

  // Uniform control flow from here: EXEC all-ones for the WMMA reductions.
  const float Ssum = wave_sum32(S);
  const float Csum = wave_sum32(C);
  const float Nsum = wave_sum32(cnt);

  if (t == 0) {
    float* pw = partial + ((size_t)s * 32 + b) * 3;
    pw[0] = Ssum;
    pw[1] = Csum;
    pw[2] = Nsum;
  }
}

// Single wave: fold the 32 batch partials per scale, combine the 3 scales.
__global__ __launch_bounds__(32) void yolo_final(const float* __restrict__ partial,
                                                 float* __restrict__ outv) {
  const int l = threadIdx.x;  // lane == batch index
  float total = 0.0f;
#pragma unroll
  for (int s = 0; s < 3; ++s) {
    const float* pw = partial + ((size_t)s * 32 + l) * 3;
    const float Ss = wave_sum32(pw[0]);
    const float Cs = wave_sum32(pw[1]);
    const float Ns = wave_sum32(pw[2]);
    const float cnt = fmaxf(Ns, 1.0f);
    total += Ss / cnt + Cs / (cnt * 80.0f);
  }
  if (l == 0) outv[0] = total;
}

extern "C" void kernel_launch(void* const* d_in, const int* in_sizes, int n_in,
                              void* d_out, int out_size, void* d_ws, size_t ws_size,
                              hipStream_t stream) {
  (void)in_sizes; (void)n_in; (void)out_size; (void)ws_size;
  const float* out0 = (const float*)d_in[0];
  const float* out1 = (const float*)d_in[1];
  const float* out2 = (const float*)d_in[2];
  const float* anc0 = (const float*)d_in[3];
  const float* anc1 = (const float*)d_in[4];
  const float* anc2 = (const float*)d_in[5];
  const float* tgts = (const float*)d_in[6];

  float* partial = (float*)d_ws;  // 3 scales * 32 batches * 3 floats = 1152 B

  dim3 grid(32, 3);
  yolo_scale_partial<<<grid, 32, 0, stream>>>(out0, out1, out2, anc0, anc1, anc2,
                                              tgts, partial);
  yolo_final<<<1, 32, 0, stream>>>(partial, (float*)d_out);
}